// VCRenderBatch_34789235097703
// MI455X (gfx1250) — compile-verified
//
#include <hip/hip_runtime.h>
#include <hip/hip_bf16.h>
#include <stdint.h>

#define H_ 64
#define W_ 64
#define B_ 8
#define P_ 512
#define F_ 512
#define REC 24
#define KAPPA (1000.0f * 1000.0f / 7000.0f)
#define EPS_ 1e-10f
#define EXPAND_ 0.02f

typedef __attribute__((ext_vector_type(2))) float v2f;
typedef __attribute__((ext_vector_type(8))) float v8f;

// ---------------------------------------------------------------------------
// Kernel 1: pcam = (verts - pos) @ rot^T  via V_WMMA_F32_16X16X4_F32.
// A (16x4): lanes 0-15 hold rows m, VGPR0=K0, VGPR1=K1; lanes 16-31 VGPR0=K2,
// VGPR1=K3 (ISA 7.12.2). B (4x16) mirrored: lane l holds column n=l&15 of
// rows K = 2*(l>>4)+vgpr. D (16x16 f32): VGPR r, lanes 0-15 -> M=r,
// lanes 16-31 -> M=r+8, N=lane&15.
// Grid: B*8 single-wave blocks; each wave does 4 row-tiles of 16.
// Loads are branchless: per-lane computed offsets, cndmask selects.
// ---------------------------------------------------------------------------
__global__ __launch_bounds__(32) void vertex_transform_wmma(
    const float* __restrict__ verts, const float* __restrict__ rot,
    const float* __restrict__ pos, float* __restrict__ pcam) {
  const int b = blockIdx.x >> 3;
  const int tgrp = blockIdx.x & 7;
  const int lane = threadIdx.x;
  const int n = lane & 15;
  const int khalf = lane >> 4;
  const float* vb = verts + b * P_ * 3;
  const float* rb = rot + b * 9;
  const float px = pos[b * 3 + 0];
  const float py = pos[b * 3 + 1];
  const float pz = pos[b * 3 + 2];
  const float sub0 = (khalf == 0) ? px : pz;   // branchless select

  // B[k][n] = rot[n][k]; k = 2*khalf (+1 for second VGPR); pad k=3 / n>=3 -> 0
  const float r0 = (n < 3) ? rb[n * 3 + 2 * khalf] : 0.0f;
  const float r1 = (n < 3) ? rb[n * 3 + 1] : 0.0f;
  v2f bf;
  bf.x = r0;
  bf.y = (khalf == 0) ? r1 : 0.0f;

#pragma unroll
  for (int i = 0; i < 4; ++i) {
    const int t = tgrp * 4 + i;
    const int row = t * 16 + n;
    // unconditional loads at per-lane offsets (no divergent branches)
    const float v0 = vb[row * 3 + 2 * khalf];  // K0 (x) or K2 (z)
    const float v1 = vb[row * 3 + 1];          // K1 (y), unused in upper half
    v2f af;
    af.x = v0 - sub0;
    af.y = (khalf == 0) ? (v1 - py) : 0.0f;
    v8f c = {};
    v8f d = __builtin_amdgcn_wmma_f32_16x16x4_f32(
        /*neg_a=*/false, af, /*neg_b=*/false, bf,
        /*c_mod=*/(short)0, c, /*reuse_a=*/false, /*reuse_b=*/false);
    if (n < 3) {
      float* out = pcam + (b * P_ + t * 16 + 8 * khalf) * 3 + n;
#pragma unroll
      for (int r = 0; r < 8; ++r) out[r * 3] = d[r];
    }
  }
}

// ---------------------------------------------------------------------------
// Kernel 2: per-face precompute. Emits normal1 output and a 24-float record:
// [0..2]  A0,B0,C0   (w0 = A0*x + B0*y + C0, division by det_safe folded in)
// [3..5]  A1,B1,C1
// [6..8]  dz0, dz1, z2   (z = z2 + w0*dz0 + w1*dz1)
// [9]     validf (normalz>0 && |det|>EPS)
// [10..15] ax,ay,bx,by,cx,cy
// [16..18] 1/(|ab|^2+eps), 1/(|bc|^2+eps), 1/(|ca|^2+eps)
// [19..22] xmin,xmax,ymin,ymax (expanded)
// [23]    pad
// ---------------------------------------------------------------------------
__global__ __launch_bounds__(256) void face_prep(
    const float* __restrict__ pcam, const int* __restrict__ faces,
    const float* __restrict__ proj, float* __restrict__ rec,
    float* __restrict__ normal_out) {
  const int gid = blockIdx.x * blockDim.x + threadIdx.x;
  if (gid >= B_ * F_) return;
  const int b = gid / F_;
  const int* fi = faces + gid * 3;
  const float* pb = pcam + b * P_ * 3;
  const float pr0 = proj[0], pr1 = proj[1], pr2 = proj[2];

  const int i0 = fi[0], i1 = fi[1], i2 = fi[2];
  const float t0x = pb[i0 * 3 + 0], t0y = pb[i0 * 3 + 1], t0z = pb[i0 * 3 + 2];
  const float t1x = pb[i1 * 3 + 0], t1y = pb[i1 * 3 + 1], t1z = pb[i1 * 3 + 2];
  const float t2x = pb[i2 * 3 + 0], t2y = pb[i2 * 3 + 1], t2z = pb[i2 * 3 + 2];

  // projection: xy = (p*proj)[:2] / (p.z*proj.z)
  const float d0 = t0z * pr2, d1 = t1z * pr2, d2v = t2z * pr2;
  const float ax = t0x * pr0 / d0, ay = t0y * pr1 / d0;
  const float bx = t1x * pr0 / d1, by = t1y * pr1 / d1;
  const float cx = t2x * pr0 / d2v, cy = t2y * pr1 / d2v;

  // normal = cross(t1-t0, t2-t0)
  const float e1x = t1x - t0x, e1y = t1y - t0y, e1z = t1z - t0z;
  const float e2x = t2x - t0x, e2y = t2y - t0y, e2z = t2z - t0z;
  const float nx = e1y * e2z - e1z * e2y;
  const float ny = e1z * e2x - e1x * e2z;
  const float nz = e1x * e2y - e1y * e2x;
  const float ninv = 1.0f / (sqrtf(nx * nx + ny * ny + nz * nz) + 1e-12f);
  normal_out[gid * 3 + 0] = nx * ninv;
  normal_out[gid * 3 + 1] = ny * ninv;
  normal_out[gid * 3 + 2] = nz * ninv;

  const float det = (by - cy) * (ax - cx) + (cx - bx) * (ay - cy);
  const float adet = fabsf(det);
  const float det_safe = (adet < EPS_) ? EPS_ : det;
  const float ids = 1.0f / det_safe;
  const float A0 = (by - cy) * ids, B0 = (cx - bx) * ids;
  const float C0 = -(A0 * cx + B0 * cy);
  const float A1 = (cy - ay) * ids, B1 = (ax - cx) * ids;
  const float C1 = -(A1 * cx + B1 * cy);
  const float validf = ((nz > 0.0f) && (adet > EPS_)) ? 1.0f : 0.0f;

  const float vabx = bx - ax, vaby = by - ay;
  const float vbcx = cx - bx, vbcy = cy - by;
  const float vcax = ax - cx, vcay = ay - cy;
  const float inv_ab = 1.0f / (vabx * vabx + vaby * vaby + 1e-12f);
  const float inv_bc = 1.0f / (vbcx * vbcx + vbcy * vbcy + 1e-12f);
  const float inv_ca = 1.0f / (vcax * vcax + vcay * vcay + 1e-12f);

  float* r = rec + gid * REC;
  r[0] = A0;  r[1] = B0;  r[2] = C0;
  r[3] = A1;  r[4] = B1;  r[5] = C1;
  r[6] = t0z - t2z;  r[7] = t1z - t2z;  r[8] = t2z;
  r[9] = validf;
  r[10] = ax;  r[11] = ay;  r[12] = bx;  r[13] = by;  r[14] = cx;  r[15] = cy;
  r[16] = inv_ab;  r[17] = inv_bc;  r[18] = inv_ca;
  r[19] = fminf(fminf(ax, bx), cx) - EXPAND_;
  r[20] = fmaxf(fmaxf(ax, bx), cx) + EXPAND_;
  r[21] = fminf(fminf(ay, by), cy) - EXPAND_;
  r[22] = fmaxf(fmaxf(ay, by), cy) + EXPAND_;
  r[23] = 0.0f;
}

__device__ __forceinline__ float seg_d2(float x, float y, float px, float py,
                                        float qx, float qy, float inv) {
  const float vx = qx - px, vy = qy - py;
  float t = ((x - px) * vx + (y - py) * vy) * inv;
  t = fminf(fmaxf(t, 0.0f), 1.0f);
  const float dx = x - fmaf(t, vx, px);
  const float dy = y - fmaf(t, vy, py);
  return dx * dx + dy * dy;
}

// ---------------------------------------------------------------------------
// Kernel 3: pixel loop. grid = B*16 (256-pixel tiles), 256 threads.
// Stages the batch's 512x24-float face table (48 KB) to LDS via CDNA5 async
// global->LDS loads, then each thread owns one pixel and streams all faces.
// The soft-coverage term (3x seg_d2 + exp) runs only for faces whose expanded
// bbox covers the pixel -> wave-level execz skip for the common case.
// ---------------------------------------------------------------------------
__global__ __launch_bounds__(256) void pixel_kernel(
    const float* __restrict__ rec, const int* __restrict__ faces,
    const float* __restrict__ colors, float* __restrict__ out) {
  __shared__ float srec[F_ * REC];  // 49152 bytes

  const int b = blockIdx.x >> 4;
  const int tile = blockIdx.x & 15;
  const int tid = threadIdx.x;

  // Async stage: 48 KB = 256 threads * 12 * b128.
  const float* src = rec + (size_t)b * F_ * REC;
  const uint32_t lds_base = (uint32_t)(uintptr_t)(&srec[0]);
#pragma unroll
  for (int i = 0; i < 12; ++i) {
    const uint32_t off = (uint32_t)(tid + i * 256) * 16u;
    const uint32_t ldsa = lds_base + off;
    asm volatile("global_load_async_to_lds_b128 %0, %1, %2"
                 :: "v"(ldsa), "v"(off), "s"(src)
                 : "memory");
  }
  asm volatile("s_wait_asynccnt 0" ::: "memory");
  __syncthreads();

  const int p = tile * 256 + tid;
  const int iy = p >> 6, ix = p & 63;
  const float x = (2.0f * (float)ix + 1.0f - (float)W_) / (float)W_;
  const float y = ((float)H_ - 2.0f * (float)iy - 1.0f) / (float)H_;

  float bestz = -1e10f;
  int best = 0;
  int anyv = 0;
  float prod = 1.0f;

#pragma unroll 2
  for (int f = 0; f < F_; ++f) {
    const float* r = &srec[f * REC];
    const float w0 = fmaf(r[0], x, fmaf(r[1], y, r[2]));
    const float w1 = fmaf(r[3], x, fmaf(r[4], y, r[5]));
    const float w2 = 1.0f - w0 - w1;
    const bool covered = (w0 >= 0.0f) && (w1 >= 0.0f) && (w2 >= 0.0f);
    const float z = fmaf(w0, r[6], fmaf(w1, r[7], r[8]));
    const bool valid = covered && (r[9] != 0.0f);
    if (valid) {
      anyv = 1;
      if (z > bestz) { bestz = z; best = f; }  // strict >: first-max, like argmax
    }
    // soft coverage term: contrib==0 outside the expanded bbox, so skip it
    const bool inb = (x >= r[19]) && (x <= r[20]) && (y >= r[21]) && (y <= r[22]);
    if (inb) {
      float d2 = 0.0f;
      if (!covered) {
        const float ax = r[10], ay = r[11], bx = r[12], by = r[13];
        const float cx = r[14], cy = r[15];
        d2 = fminf(fminf(seg_d2(x, y, ax, ay, bx, by, r[16]),
                         seg_d2(x, y, bx, by, cx, cy, r[17])),
                   seg_d2(x, y, cx, cy, ax, ay, r[18]));
      }
      prod *= (1.0f - __expf(-d2 * KAPPA));
    }
  }

  // Finalize: winning-face barycentrics + color gather.
  const float* rb = &srec[best * REC];
  const float g0 = fmaf(rb[0], x, fmaf(rb[1], y, rb[2]));
  const float g1 = fmaf(rb[3], x, fmaf(rb[4], y, rb[5]));
  const float g2 = 1.0f - g0 - g1;

  float rr = 0.0f, gg = 0.0f, bb = 0.0f, hm = 0.0f;
  if (anyv) {
    const int* fi = faces + (b * F_ + best) * 3;
    const float* cb = colors + b * P_ * 3;
    const float* c0 = cb + fi[0] * 3;
    const float* c1 = cb + fi[1] * 3;
    const float* c2 = cb + fi[2] * 3;
    rr = g0 * c0[0] + g1 * c1[0] + g2 * c2[0];
    gg = g0 * c0[1] + g1 * c1[1] + g2 * c2[1];
    bb = g0 * c0[2] + g1 * c1[2] + g2 * c2[2];
    hm = g0 + g1 + g2;
  }

  const int pix = b * (H_ * W_) + p;
  out[pix * 3 + 0] = rr;                 // imrender: [0, 98304)
  out[pix * 3 + 1] = gg;
  out[pix * 3 + 2] = bb;
  out[98304 + pix] = 1.0f - prod;        // improb:   [98304, 131072)
  out[143360 + pix] = hm;                // hardmask: [143360, 176128)
}

// ---------------------------------------------------------------------------
extern "C" void kernel_launch(void* const* d_in, const int* in_sizes, int n_in,
                              void* d_out, int out_size, void* d_ws,
                              size_t ws_size, hipStream_t stream) {
  const float* verts  = (const float*)d_in[0];
  const int*   faces  = (const int*)d_in[1];
  const float* rot    = (const float*)d_in[2];
  const float* pos    = (const float*)d_in[3];
  const float* proj   = (const float*)d_in[4];
  const float* colors = (const float*)d_in[5];
  float* out = (float*)d_out;

  float* pcam = (float*)d_ws;            // B*P*3 floats
  float* rec  = pcam + B_ * P_ * 3;      // B*F*REC floats
  float* normal_out = out + 131072;      // normal1: [131072, 143360)

  vertex_transform_wmma<<<B_ * 8, 32, 0, stream>>>(verts, rot, pos, pcam);
  face_prep<<<(B_ * F_ + 255) / 256, 256, 0, stream>>>(pcam, faces, proj, rec,
                                                       normal_out);
  pixel_kernel<<<B_ * 16, 256, 0, stream>>>(rec, faces, colors, out);
}